// GNLoss_84000970375606
// MI455X (gfx1250) — compile-verified
//
#include <hip/hip_runtime.h>
#include <hip/hip_bf16.h>
#include <math.h>

typedef float v2f __attribute__((ext_vector_type(2)));
typedef float v8f __attribute__((ext_vector_type(8)));

#define C_CH 32
#define N_PTS 2048
#define BN_TOT 8192

__device__ __forceinline__ int imin(int a, int b) { return a < b ? a : b; }
__device__ __forceinline__ int imax(int a, int b) { return a > b ? a : b; }

// ---- simple bilinear corner info (clip + floor, per reference) ----
struct Corner { int r0, r1, c0, c1; float wx, wy; };

__device__ __forceinline__ void mkCorner(float px, float py, int W, Corner& cr) {
  float fx = fminf(fmaxf(px, 0.f), (float)(W - 1));
  float fy = fminf(fmaxf(py, 0.f), (float)(W - 1));
  float x0 = floorf(fx), y0 = floorf(fy);
  cr.wx = fx - x0; cr.wy = fy - y0;
  int xi = (int)x0, yi = (int)y0;
  cr.c0 = xi; cr.c1 = imin(xi + 1, W - 1);
  cr.r0 = yi * W; cr.r1 = imin(yi + 1, W - 1) * W;
}

__device__ __forceinline__ float bilin(const float* __restrict__ p, const Corner& cr) {
  float f00 = p[cr.r0 + cr.c0], f01 = p[cr.r0 + cr.c1];
  float f10 = p[cr.r1 + cr.c0], f11 = p[cr.r1 + cr.c1];
  float wx = cr.wx, wy = cr.wy;
  return f00 * (1.f - wx) * (1.f - wy) + f01 * wx * (1.f - wy)
       + f10 * (1.f - wx) * wy + f11 * wx * wy;
}

// ---- GN sample: value + on-the-fly central-difference Jacobian (edge-replicated) ----
struct GnCorner { int rA, rB, rC, rD, cA, cB, cC, cD; float wx, wy; bool xe, ye; };

__device__ __forceinline__ void mkGn(float px, float py, int W, GnCorner& g) {
  float fx = fminf(fmaxf(px, 0.f), (float)(W - 1));
  float fy = fminf(fmaxf(py, 0.f), (float)(W - 1));
  float x0 = floorf(fx), y0 = floorf(fy);
  g.wx = fx - x0; g.wy = fy - y0;
  int xi = (int)x0, yi = (int)y0;
  int x1 = imin(xi + 1, W - 1), y1 = imin(yi + 1, W - 1);
  g.cA = imax(xi - 1, 0); g.cB = xi; g.cC = x1; g.cD = imin(x1 + 1, W - 1);
  g.rA = imax(yi - 1, 0) * W; g.rB = yi * W; g.rC = y1 * W; g.rD = imin(y1 + 1, W - 1) * W;
  g.xe = xi < W - 1; g.ye = yi < W - 1;
}

__device__ __forceinline__ void sampleGn(const float* __restrict__ p, const GnCorner& g,
                                         float& fs, float& jx, float& jy) {
  float fBA = p[g.rB + g.cA], fBB = p[g.rB + g.cB], fBC = p[g.rB + g.cC], fBD = p[g.rB + g.cD];
  float fCA = p[g.rC + g.cA], fCB = p[g.rC + g.cB], fCC = p[g.rC + g.cC], fCD = p[g.rC + g.cD];
  float fAB = p[g.rA + g.cB], fAC = p[g.rA + g.cC];
  float fDB = p[g.rD + g.cB], fDC = p[g.rD + g.cC];
  float wx = g.wx, wy = g.wy;
  float w00 = (1.f - wx) * (1.f - wy), w01 = wx * (1.f - wy);
  float w10 = (1.f - wx) * wy,         w11 = wx * wy;
  fs = fBB * w00 + fBC * w01 + fCB * w10 + fCC * w11;
  float gx00 = 0.5f * ((g.xe ? fBC : fBB) - fBA);
  float gx01 = 0.5f * (fBD - (g.xe ? fBB : fBA));
  float gx10 = 0.5f * ((g.xe ? fCC : fCB) - fCA);
  float gx11 = 0.5f * (fCD - (g.xe ? fCB : fCA));
  jx = gx00 * w00 + gx01 * w01 + gx10 * w10 + gx11 * w11;
  float gy00 = 0.5f * ((g.ye ? fCB : fBB) - fAB);
  float gy01 = 0.5f * ((g.ye ? fCC : fBC) - fAC);
  float gy10 = 0.5f * (fDB - (g.ye ? fBB : fAB));
  float gy11 = 0.5f * (fDC - (g.ye ? fBC : fAC));
  jy = gy00 * w00 + gy01 * w01 + gy10 * w10 + gy11 * w11;
}

// One wave (32 lanes) owns 16 points; lane l handles channels {4k+2*(l/16), +1}
// of point l%16 per K-chunk k, matching the WMMA f32 16x16x4 A/B layout.
__global__ __launch_bounds__(256) void gn_level_kernel(
    const float* __restrict__ fa, const float* __restrict__ fb,
    const float* __restrict__ noise,
    const float* __restrict__ ma, const float* __restrict__ mb,
    const float* __restrict__ ng,
    int W, float inv_s, float* __restrict__ ws) {
  const int lane = threadIdx.x & 31;
  const int h = lane >> 4;
  const int wv = blockIdx.x * (blockDim.x >> 5) + (threadIdx.x >> 5);
  const int p = wv * 16 + (lane & 15);            // 0 .. 8191
  const int b = p >> 11;                          // N = 2048
  const int HW = W * W;

  const float2 mav = ((const float2*)ma)[p];
  const float2 mbv = ((const float2*)mb)[p];
  const float2 ngv = ((const float2*)ng)[p];
  const float2 nsv = ((const float2*)noise)[p];
  const float ubx = mbv.x * inv_s, uby = mbv.y * inv_s;

  Corner ca, cb, cn; GnCorner gg;
  mkCorner(mav.x * inv_s, mav.y * inv_s, W, ca);
  mkCorner(ubx, uby, W, cb);
  mkCorner(ngv.x * inv_s, ngv.y * inv_s, W, cn);
  mkGn(nsv.x + ubx, nsv.y + uby, W, gg);

  v8f accA = {}, accDD = {}, accB = {}, accBX = {}, accBY = {}, accP = {}, accN = {};
  const int base = b * C_CH;

  for (int kc = 0; kc < 8; ++kc) {
    float jx[2], jy[2], rr[2], da[2], dnv[2];
#pragma unroll
    for (int j = 0; j < 2; ++j) {
      int c = 4 * kc + 2 * h + j;
      const float* pa = fa + (size_t)(base + c) * HW;
      const float* pb = fb + (size_t)(base + c) * HW;
      float ft = bilin(pa, ca);               // == triplet anchor sample
      float fp = bilin(pb, cb);
      float fn = bilin(pb, cn);
      float fs, jxx, jyy;
      sampleGn(pb, gg, fs, jxx, jyy);
      jx[j] = jxx; jy[j] = jyy; rr[j] = fs - ft;
      da[j] = ft - fp; dnv[j] = ft - fn;
    }
    v2f oJx = {jx[0], jx[1]}, oJy = {jy[0], jy[1]}, oR = {rr[0], rr[1]};
    v2f oDa = {da[0], da[1]}, oDn = {dnv[0], dnv[1]};
    // D = A*B + C with B registers == A layout -> D = V * V^T (Gram); diag = dot products
    accA  = __builtin_amdgcn_wmma_f32_16x16x4_f32(false, oJx, false, oJx, (short)0, accA,  false, false);
    accDD = __builtin_amdgcn_wmma_f32_16x16x4_f32(false, oJy, false, oJy, (short)0, accDD, false, false);
    accB  = __builtin_amdgcn_wmma_f32_16x16x4_f32(false, oJx, false, oJy, (short)0, accB,  false, false);
    accBX = __builtin_amdgcn_wmma_f32_16x16x4_f32(false, oJx, false, oR,  (short)0, accBX, false, false);
    accBY = __builtin_amdgcn_wmma_f32_16x16x4_f32(false, oJy, false, oR,  (short)0, accBY, false, false);
    accP  = __builtin_amdgcn_wmma_f32_16x16x4_f32(false, oDa, false, oDa, (short)0, accP,  false, false);
    accN  = __builtin_amdgcn_wmma_f32_16x16x4_f32(false, oDn, false, oDn, (short)0, accN,  false, false);
  }

  // Diagonal (m,m): VGPR m%8 at lane m (m<8) or lane m+16 (m>=8) — that lane owns point m.
  float lq = 0.f, lld = 0.f, lrelu = 0.f;
#pragma unroll
  for (int v = 0; v < 8; ++v) {
    if ((lane == v) || (lane == v + 24)) {
      float a  = accA[v]  + 1e-9f;
      float d  = accDD[v] + 1e-9f;
      float bb = accB[v];
      float det = a * d - bb * bb;
      float bx = accBX[v], by = accBY[v];
      float ix = (d * bx - bb * by) / det;
      float iy = (a * by - bb * bx) / det;
      float dux = ix - nsv.x, duy = iy - nsv.y;   // du = -noise + Hinv*J^T r
      lq  += a * dux * dux + 2.f * bb * dux * duy + d * duy * duy;
      lld += logf(det);
      lrelu += fmaxf(sqrtf(accP[v]) - sqrtf(accN[v]) + 1.0f, 0.f);
    }
  }
  for (int off = 16; off > 0; off >>= 1) {
    lq    += __shfl_xor(lq, off, 32);
    lld   += __shfl_xor(lld, off, 32);
    lrelu += __shfl_xor(lrelu, off, 32);
  }
  if (lane == 0) {
    atomicAdd(&ws[0], lrelu);
    atomicAdd(&ws[1], lq);
    atomicAdd(&ws[2], lld);
  }
}

__global__ void gn_init_kernel(float* ws) {
  if (threadIdx.x < 3) ws[threadIdx.x] = 0.f;
}

__global__ void gn_final_kernel(const float* __restrict__ ws, float* __restrict__ out) {
  if (threadIdx.x == 0) {
    float tl = 100.f * ws[0] / (float)BN_TOT;                  // CONTR_LAMDA * mean
    float l2pi = logf(2.f * 3.14159265358979323846f);
    // gl = 0.3 * [ 0.5*Σq + Σ_levels (2/7)(BN*log2pi) - (1/7)*Σ log det ]
    float gl = 0.3f * (0.5f * ws[1] - ws[2] / 7.f + (8.f / 7.f) * (float)BN_TOT * l2pi);
    out[0] = tl + gl; out[1] = tl; out[2] = gl;
  }
}

extern "C" void kernel_launch(void* const* d_in, const int* in_sizes, int n_in,
                              void* d_out, int out_size, void* d_ws, size_t ws_size,
                              hipStream_t stream) {
  // setup_inputs dict order: fa0,fb0,noise0, fa1,fb1,noise1, fa2,fb2,noise2,
  //                          fa3,fb3,noise3, ma, mb, neg, epoch
  const float* ma  = (const float*)d_in[12];
  const float* mb  = (const float*)d_in[13];
  const float* ng  = (const float*)d_in[14];
  float* ws  = (float*)d_ws;
  float* out = (float*)d_out;

  gn_init_kernel<<<1, 32, 0, stream>>>(ws);

  for (int i = 0; i < 4; ++i) {
    const float* fa    = (const float*)d_in[3 * i + 0];
    const float* fb    = (const float*)d_in[3 * i + 1];
    const float* noise = (const float*)d_in[3 * i + 2];
    int W = 64 << i;
    float inv_s = 1.0f / (float)(16 >> i);   // s = 2^(3-i)*2 = 16,8,4,2
    // 8192 points / 16 per wave = 512 waves; 8 waves per 256-thread block
    gn_level_kernel<<<64, 256, 0, stream>>>(fa, fb, noise, ma, mb, ng, W, inv_s, ws);
  }

  gn_final_kernel<<<1, 32, 0, stream>>>(ws, out);
}